// GNNLayer_2963527434325
// MI455X (gfx1250) — compile-verified
//
#include <hip/hip_runtime.h>

#define DIM 128  // DIM_IN == DIM_OUT == 128

typedef __attribute__((ext_vector_type(2))) float v2f;
typedef __attribute__((ext_vector_type(8))) float v8f;

// -----------------------------------------------------------------------------
// Kernel 1: h = x @ W^T using V_WMMA_F32_16X16X4_F32 (exact fp32 matrix path).
// One wave32 per 16x16 output tile. Block = 256 threads = 8 waves; the 8 waves
// of a block share the same 16 rows of x and each take one of the 8 N-tiles.
// K-loop: 32 x (load float2 A, load float2 B, wmma).
//
// A (16x4 f32) per-lane layout: M = lane&15; lane<16 -> K={0,1}, lane>=16 -> K={2,3}.
// B (4x16) is W^T, i.e. B[k][n] = W[n][k]; mirrored layout -> identical float2
// fetch pattern from W's row n = tileN*16 + (lane&15).
// C/D (16x16 f32, 8 VGPRs): lane<16 -> M=i, lane>=16 -> M=i+8; N = lane&15.
// -----------------------------------------------------------------------------
__global__ __launch_bounds__(256) void gnn_gemm_wmma_f32(
    const float* __restrict__ x, const float* __restrict__ W,
    float* __restrict__ h)
{
    const int lane   = threadIdx.x & 31;
    const int tileN  = threadIdx.x >> 5;   // 0..7
    const int tileM  = blockIdx.x;         // n_nodes/16 tiles (100000/16 = 6250)
    const int mIdx   = lane & 15;
    const int kOff   = (lane >> 4) << 1;   // lanes 0-15 -> K base 0, lanes 16-31 -> K base 2

    const float* __restrict__ xA = x + (size_t)(tileM * 16 + mIdx) * DIM + kOff;
    const float* __restrict__ wB = W + (size_t)(tileN * 16 + mIdx) * DIM + kOff;

    v8f c = {};
#pragma unroll
    for (int k = 0; k < DIM; k += 4) {
        v2f a = *(const v2f*)(xA + k);
        v2f b = *(const v2f*)(wB + k);
        // 8 args: (neg_a, A, neg_b, B, c_mod, C, reuse_a, reuse_b)
        c = __builtin_amdgcn_wmma_f32_16x16x4_f32(
                false, a, false, b, (short)0, c, false, false);
    }

    // Writeback: lane<16 covers M=0..7, lane>=16 covers M=8..15 of the tile.
    const int mBase = tileM * 16 + ((lane >> 4) << 3);
    float* __restrict__ hp = h + (size_t)mBase * DIM + tileN * 16 + mIdx;
#pragma unroll
    for (int i = 0; i < 8; ++i)
        hp[(size_t)i * DIM] = c[i];
}

// -----------------------------------------------------------------------------
// Kernel 2: zero d_out (harness poisons it; atomics below need a clean base).
// -----------------------------------------------------------------------------
__global__ __launch_bounds__(256) void gnn_zero_out(float4* __restrict__ out, int n4)
{
    int i = blockIdx.x * blockDim.x + threadIdx.x;
    if (i < n4) out[i] = make_float4(0.f, 0.f, 0.f, 0.f);
}

// -----------------------------------------------------------------------------
// Kernel 3: SpMM scatter. One wave32 per edge; lane L moves float4 covering
// features 4L..4L+3 (32 lanes x 16 B = full 128-float row, coalesced).
// h (51 MB) and out (51 MB) both fit in the 192 MB L2, so the gathers and the
// non-returning global_atomic_add_f32 traffic run at L2 bandwidth.
// -----------------------------------------------------------------------------
__global__ __launch_bounds__(256) void gnn_spmm_scatter(
    const float* __restrict__ h,
    const int*   __restrict__ erow,
    const int*   __restrict__ ecol,
    const float* __restrict__ eval,
    float*       __restrict__ out,
    int n_edges)
{
    int wave = (int)(((size_t)blockIdx.x * blockDim.x + threadIdx.x) >> 5);
    int lane = threadIdx.x & 31;
    if (wave >= n_edges) return;

    // wave id is uniform across the wave -> force scalar loads of edge data
    int e = __builtin_amdgcn_readfirstlane(wave);
    int   r = erow[e];
    int   c = ecol[e];
    float v = eval[e];

    const float4* __restrict__ hp = (const float4*)(h + (size_t)c * DIM);
    float4 m = hp[lane];

    float* __restrict__ op = out + (size_t)r * DIM + lane * 4;
    atomicAdd(op + 0, v * m.x);
    atomicAdd(op + 1, v * m.y);
    atomicAdd(op + 2, v * m.z);
    atomicAdd(op + 3, v * m.w);
}

// -----------------------------------------------------------------------------
// Launcher
// -----------------------------------------------------------------------------
extern "C" void kernel_launch(void* const* d_in, const int* in_sizes, int n_in,
                              void* d_out, int out_size, void* d_ws, size_t ws_size,
                              hipStream_t stream)
{
    const float* x    = (const float*)d_in[0];   // [N, 128]
    const float* W    = (const float*)d_in[1];   // [128, 128]
    const int*   erow = (const int*)  d_in[2];   // [E]
    const int*   ecol = (const int*)  d_in[3];   // [E]
    const float* eval = (const float*)d_in[4];   // [E]
    float*       out  = (float*)d_out;           // [N, 128]
    float*       h    = (float*)d_ws;            // scratch: N*128 floats (51.2 MB)

    const int n_nodes = in_sizes[0] / DIM;       // 100000
    const int n_edges = in_sizes[2];             // 1600000

    // GEMM: one block per 16-row stripe, 8 waves cover the 8 N-tiles.
    const int mTiles = n_nodes / 16;             // 6250 (N divides exactly)
    gnn_gemm_wmma_f32<<<mTiles, 256, 0, stream>>>(x, W, h);

    // Zero the output accumulator.
    const int n4 = out_size / 4;
    gnn_zero_out<<<(n4 + 255) / 256, 256, 0, stream>>>((float4*)out, n4);

    // Scatter: 8 edges (waves) per 256-thread block.
    const int blocks = (n_edges + 7) / 8;
    gnn_spmm_scatter<<<blocks, 256, 0, stream>>>(h, erow, ecol, eval, out, n_edges);
}